// StochasticGCN_9723805958348
// MI455X (gfx1250) — compile-verified
//
#include <hip/hip_runtime.h>
#include <math.h>

typedef float v2f __attribute__((ext_vector_type(2)));
typedef float v8f __attribute__((ext_vector_type(8)));

static constexpr int cN0 = 200000, cN1 = 50000, cN2 = 10000;
static constexpr int IN_DIM = 256, H1 = 256, H2 = 256, EMB = 128;

// ---------------------------------------------------------------------------
// Degree counting: one thread per edge, float atomic increments.
// ---------------------------------------------------------------------------
__global__ void deg_count_kernel(const int* __restrict__ idx,
                                 float* __restrict__ deg, int n) {
    int i = blockIdx.x * blockDim.x + threadIdx.x;
    if (i < n) atomicAdd(&deg[idx[i]], 1.0f);
}

// ---------------------------------------------------------------------------
// Edge scatter: agg[dst] += X[src] * rsqrt(max(out_deg[src],1)).
// 64 threads per edge, float4 per thread (64*4 = 256 columns).
// Destination buffers are small (<=51MB) -> atomics resolve in L2.
// ---------------------------------------------------------------------------
__global__ __launch_bounds__(256) void scatter_edges_kernel(
    const float* __restrict__ X,       // [n_src, 256]
    const int* __restrict__ src,
    const int* __restrict__ dst,
    const float* __restrict__ deg_out, // [n_src]
    float* __restrict__ agg,           // [n_dst, 256], pre-zeroed
    int nedges) {
    int t = blockIdx.x * 256 + threadIdx.x;
    int e = t >> 6;
    if (e >= nedges) return;
    int lane = t & 63;
    int s = src[e];
    int d = dst[e];
    float scale = rsqrtf(fmaxf(deg_out[s], 1.0f));
    float4 v = reinterpret_cast<const float4*>(X + (size_t)s * 256)[lane];
    float* out = agg + (size_t)d * 256 + lane * 4;
    atomicAdd(out + 0, v.x * scale);
    atomicAdd(out + 1, v.y * scale);
    atomicAdd(out + 2, v.z * scale);
    atomicAdd(out + 3, v.w * scale);
}

// ---------------------------------------------------------------------------
// Fused GEMM + row-scale + bias using V_WMMA_F32_16X16X4_F32 (fp32, exact
// precision class of the reference). C = rowscale(deg) * (A @ W) + bias.
//   A: [M, 256] row-major, M % 16 == 0
//   W: [256, NOUT] row-major (<=256KB, L2-resident)
// Block = 128 threads (4 waves). One 16-row block per CTA. A staged in LDS
// with padded stride (260 floats -> 4-bank row shift, conflict-free b64
// fragment reads). Each wave owns NOUT/4 columns (NT = NOUT/64 tiles).
// HAS_DEG is a compile-time flag so the epilogue is branch-free.
// ---------------------------------------------------------------------------
template <int NOUT, bool HAS_DEG>
__global__ __launch_bounds__(128) void gemm_wmma_kernel(
    const float* __restrict__ A,
    const float* __restrict__ W,
    const float* __restrict__ bias,
    const float* __restrict__ deg,   // row scale source (unused if !HAS_DEG)
    float* __restrict__ C) {
    constexpr int LDA = 260;          // padded LDS row stride (floats)
    constexpr int NT  = NOUT / 64;    // 16x16 tiles per wave
    __shared__ float As[16 * LDA];

    const int row0 = blockIdx.x * 16;

    // Warm L2 path for W (emits global_prefetch_b8).
    __builtin_prefetch(W + (size_t)threadIdx.x * 16, 0, 1);

    // Cooperative stage of the 16x256 A block (16 rows x 64 float4 chunks).
    for (int i = threadIdx.x; i < 16 * 64; i += 128) {
        int row = i >> 6;
        int c4  = i & 63;
        *reinterpret_cast<float4*>(&As[row * LDA + c4 * 4]) =
            reinterpret_cast<const float4*>(A + (size_t)(row0 + row) * 256)[c4];
    }
    __syncthreads();

    const int wave   = threadIdx.x >> 5;   // 0..3
    const int lane   = threadIdx.x & 31;
    const int lhalf  = lane >> 4;          // 0 or 1
    const int l16    = lane & 15;
    const int koff   = lhalf * 2;          // K sub-offset handled by this half
    const int n_base = wave * (NOUT / 4);

    v8f acc[NT];
    v8f zero = {0.f, 0.f, 0.f, 0.f, 0.f, 0.f, 0.f, 0.f};
#pragma unroll
    for (int t = 0; t < NT; ++t) acc[t] = zero;

    // K = 256 in 64 steps of K=4.
    for (int k4 = 0; k4 < 256; k4 += 4) {
        v2f a;
        a.x = As[l16 * LDA + k4 + koff];
        a.y = As[l16 * LDA + k4 + koff + 1];
#pragma unroll
        for (int t = 0; t < NT; ++t) {
            int n = n_base + t * 16 + l16;
            v2f b;
            b.x = W[(size_t)(k4 + koff) * NOUT + n];
            b.y = W[(size_t)(k4 + koff + 1) * NOUT + n];
            acc[t] = __builtin_amdgcn_wmma_f32_16x16x4_f32(
                false, a, false, b, (short)0, acc[t], false, false);
        }
    }

    // Epilogue: branch-free rsqrt(max(deg,1)) row scale + bias.
    float scale[8];
#pragma unroll
    for (int r = 0; r < 8; ++r) {
        if (HAS_DEG) {
            int row = row0 + r + 8 * lhalf;
            scale[r] = rsqrtf(fmaxf(deg[row], 1.0f));
        } else {
            scale[r] = 1.0f;
        }
    }
#pragma unroll
    for (int t = 0; t < NT; ++t) {
        int n = n_base + t * 16 + l16;
        float bv = bias[n];
#pragma unroll
        for (int r = 0; r < 8; ++r) {
            int row = row0 + r + 8 * lhalf;
            float v = HAS_DEG ? acc[t][r] * scale[r] : acc[t][r];
            C[(size_t)row * NOUT + n] = v + bv;
        }
    }
}

// ---------------------------------------------------------------------------
// Host-side orchestration (graph-capture safe: only async ops on `stream`).
// ---------------------------------------------------------------------------
extern "C" void kernel_launch(void* const* d_in, const int* in_sizes, int n_in,
                              void* d_out, int out_size, void* d_ws, size_t ws_size,
                              hipStream_t stream) {
    const float* h    = (const float*)d_in[0];
    const int*   src0 = (const int*)  d_in[1];
    const int*   dst0 = (const int*)  d_in[2];
    const int*   src1 = (const int*)  d_in[3];
    const int*   dst1 = (const int*)  d_in[4];
    const float* W1   = (const float*)d_in[5];
    const float* b1   = (const float*)d_in[6];
    const float* W2   = (const float*)d_in[7];
    const float* b2   = (const float*)d_in[8];
    const float* Wp   = (const float*)d_in[9];
    const float* bp   = (const float*)d_in[10];
    float* out = (float*)d_out;

    const int E0 = in_sizes[1];
    const int E1 = in_sizes[3];

    // Workspace layout (floats). Zeroed region first, then scratch outputs.
    float* ws = (float*)d_ws;
    size_t o = 0;
    float* deg_out0 = ws + o; o += cN0;              // out-degree over src0
    float* deg_in0  = ws + o; o += cN1;              // in-degree  over dst0
    float* deg_out1 = ws + o; o += cN1;              // out-degree over src1
    float* deg_in1  = ws + o; o += cN2;              // in-degree  over dst1
    float* agg1     = ws + o; o += (size_t)cN1 * IN_DIM;
    float* agg2     = ws + o; o += (size_t)cN2 * H1;
    size_t zero_floats = o;                          // everything above -> 0
    float* h1       = ws + o; o += (size_t)cN1 * H1;
    float* h2       = ws + o; o += (size_t)cN2 * H2;

    hipMemsetAsync(ws, 0, zero_floats * sizeof(float), stream);

    // Degrees.
    deg_count_kernel<<<(E0 + 255) / 256, 256, 0, stream>>>(src0, deg_out0, E0);
    deg_count_kernel<<<(E0 + 255) / 256, 256, 0, stream>>>(dst0, deg_in0, E0);
    deg_count_kernel<<<(E1 + 255) / 256, 256, 0, stream>>>(src1, deg_out1, E1);
    deg_count_kernel<<<(E1 + 255) / 256, 256, 0, stream>>>(dst1, deg_in1, E1);

    // Layer 1: aggregate-then-GEMM (linearity lets segsum commute with @W1).
    scatter_edges_kernel<<<(E0 * 64 + 255) / 256, 256, 0, stream>>>(
        h, src0, dst0, deg_out0, agg1, E0);
    gemm_wmma_kernel<H1, true><<<cN1 / 16, 128, 0, stream>>>(
        agg1, W1, b1, deg_in0, h1);

    // Layer 2.
    scatter_edges_kernel<<<(E1 * 64 + 255) / 256, 256, 0, stream>>>(
        h1, src1, dst1, deg_out1, agg2, E1);
    gemm_wmma_kernel<H2, true><<<cN2 / 16, 128, 0, stream>>>(
        agg2, W2, b2, deg_in1, h2);

    // Projection (no row scale).
    gemm_wmma_kernel<EMB, false><<<cN2 / 16, 128, 0, stream>>>(
        h2, Wp, bp, nullptr, out);
}